// MetaAttentionLayer_35785667510288
// MI455X (gfx1250) — compile-verified
//
#include <hip/hip_runtime.h>
#include <hip/hip_bf16.h>

typedef float v2f __attribute__((ext_vector_type(2)));
typedef float v8f __attribute__((ext_vector_type(8)));

#define IN_C  128
#define OUT_C 64
#define HEADS 8
#define HO    (HEADS * OUT_C)   // 512

// ---- monotonic float<->uint mapping for atomicMax on signed floats ----
__device__ __forceinline__ unsigned fmap(float f) {
    unsigned u = __float_as_uint(f);
    return (u >> 31) ? ~u : (u | 0x80000000u);
}
__device__ __forceinline__ float funmap(unsigned m) {
    unsigned u = (m >> 31) ? (m & 0x7FFFFFFFu) : ~m;
    return __uint_as_float(u);
}
__device__ __forceinline__ float lrelu(float v) { return v > 0.f ? v : 0.01f * v; }

// ---- zero output, init per-head max/sum slots ----
__global__ void k_init(float* __restrict__ out, long long total,
                       unsigned* __restrict__ maxKeys, float* __restrict__ sums) {
    long long i = (long long)blockIdx.x * blockDim.x + threadIdx.x;
    long long stride = (long long)gridDim.x * blockDim.x;
    for (; i < total; i += stride) out[i] = 0.f;
    if (blockIdx.x == 0 && threadIdx.x < HEADS) {
        maxKeys[threadIdx.x] = 0u;   // below fmap() of any real float
        sums[threadIdx.x]    = 0.f;
    }
}

// ---- domain path: 64x128 matvec + 8-way softmax, writes dw[h]/HEADS ----
__global__ void k_domain(const float* __restrict__ dc, const float* __restrict__ dW,
                         const float* __restrict__ db, const float* __restrict__ dA,
                         float* __restrict__ dwH) {
    __shared__ float proj[OUT_C];
    int o = threadIdx.x;
    float s = db[o];
    for (int i = 0; i < IN_C; ++i) s += dc[i] * dW[o * IN_C + i];
    proj[o] = s;
    __syncthreads();
    if (o == 0) {
        float logit[HEADS]; float m = -1e30f;
        for (int h = 0; h < HEADS; ++h) {
            float l = 0.f;
            for (int c = 0; c < OUT_C; ++c) l += proj[c] * dA[h * OUT_C + c];
            logit[h] = l; m = fmaxf(m, l);
        }
        float se = 0.f;
        for (int h = 0; h < HEADS; ++h) { logit[h] = __expf(logit[h] - m); se += logit[h]; }
        for (int h = 0; h < HEADS; ++h) dwH[h] = logit[h] / (se * (float)HEADS);
    }
}

// ---- fused GEMM (f32 WMMA) + per-head attention logits ----
// block = 256 thr = 8 waves; wave h computes the 16x64 strip of head h for this
// 16-row node tile. K=128 swept with V_WMMA_F32_16X16X4_F32 (K-step 4).
__global__ void __launch_bounds__(256)
k_gemm(const float* __restrict__ x, const float* __restrict__ mw,
       const float* __restrict__ matt, float* __restrict__ xt,
       float* __restrict__ ssrc, float* __restrict__ sdst, int n) {
    __shared__ float As[16][IN_C + 4];
    const int rowBase = blockIdx.x * 16;
    const bool fullTile = (rowBase + 16 <= n);

    if (fullTile) {
        // branch-free coalesced tile load (2 x b128-wide rows per thread)
        for (int i = threadIdx.x; i < 16 * IN_C; i += 256) {
            int r = i >> 7, c = i & (IN_C - 1);
            As[r][c] = x[(size_t)(rowBase + r) * IN_C + c];
        }
    } else {
        for (int i = threadIdx.x; i < 16 * IN_C; i += 256) {
            int r = i >> 7, c = i & (IN_C - 1);
            int gr = rowBase + r;
            As[r][c] = (gr < n) ? x[(size_t)gr * IN_C + c] : 0.f;
        }
    }
    __syncthreads();

    const int h    = threadIdx.x >> 5;   // wave id == head
    const int lane = threadIdx.x & 31;
    const int half = lane >> 4;          // K-group select per ISA A/B layout
    const int ln   = lane & 15;          // M (for A) / N (for B)
    const float* Wh = mw + (size_t)h * IN_C * OUT_C;  // (128,64) row-major

    v8f acc[4] = {};                     // 4 tiles of 16x16 cover OUT_C=64

    for (int kk = 0; kk < IN_C; kk += 4) {
        const int k0 = kk + half * 2;
        v2f a; a.x = As[ln][k0]; a.y = As[ln][k0 + 1];
        #pragma unroll
        for (int t = 0; t < 4; ++t) {
            v2f b;
            b.x = Wh[(size_t)k0 * OUT_C + t * 16 + ln];
            b.y = Wh[(size_t)(k0 + 1) * OUT_C + t * 16 + ln];
            acc[t] = __builtin_amdgcn_wmma_f32_16x16x4_f32(
                false, a, false, b, (short)0, acc[t], false, false);
        }
    }

    // store xt node-major [n][h][o]; C/D layout: M = r + 8*half, o = t*16 + ln.
    // Fast path: single base pointer, all 32 stores use immediate offsets.
    const int rowM0 = rowBase + 8 * half;
    float* dst = xt + (size_t)rowM0 * HO + h * OUT_C + ln;
    if (fullTile) {
        #pragma unroll
        for (int r = 0; r < 8; ++r) {
            #pragma unroll
            for (int t = 0; t < 4; ++t)
                dst[(size_t)r * HO + t * 16] = acc[t][r];
        }
    } else {
        #pragma unroll
        for (int r = 0; r < 8; ++r) {
            if (rowM0 + r < n) {
                #pragma unroll
                for (int t = 0; t < 4; ++t)
                    dst[(size_t)r * HO + t * 16] = acc[t][r];
            }
        }
    }

    // attention logits straight from accumulators (saves re-reading xt)
    const float* asrc = matt + h * (2 * OUT_C);
    const float* adst = asrc + OUT_C;
    float ps[8], pd[8];
    #pragma unroll
    for (int r = 0; r < 8; ++r) { ps[r] = 0.f; pd[r] = 0.f; }
    #pragma unroll
    for (int t = 0; t < 4; ++t) {
        float av = asrc[t * 16 + ln];
        float dv = adst[t * 16 + ln];
        #pragma unroll
        for (int r = 0; r < 8; ++r) { ps[r] += acc[t][r] * av; pd[r] += acc[t][r] * dv; }
    }
    #pragma unroll
    for (int m = 8; m >= 1; m >>= 1) {   // butterfly across the 16 N-lanes
        #pragma unroll
        for (int r = 0; r < 8; ++r) {
            ps[r] += __shfl_xor(ps[r], m, 32);
            pd[r] += __shfl_xor(pd[r], m, 32);
        }
    }
    if (ln == 0) {
        float* psrc = ssrc + (size_t)rowM0 * HEADS + h;
        float* pdst = sdst + (size_t)rowM0 * HEADS + h;
        if (fullTile) {
            #pragma unroll
            for (int r = 0; r < 8; ++r) {
                psrc[(size_t)r * HEADS] = ps[r];
                pdst[(size_t)r * HEADS] = pd[r];
            }
        } else {
            #pragma unroll
            for (int r = 0; r < 8; ++r) {
                if (rowM0 + r < n) {
                    psrc[(size_t)r * HEADS] = ps[r];
                    pdst[(size_t)r * HEADS] = pd[r];
                }
            }
        }
    }
}

// ---- pass 1: per-head global max of leaky-relu scores over all edges ----
__global__ void k_edge_max(const int* __restrict__ ei, const float* __restrict__ ssrc,
                           const float* __restrict__ sdst, unsigned* __restrict__ maxKeys,
                           int E) {
    __shared__ unsigned sm[HEADS];
    if (threadIdx.x < HEADS) sm[threadIdx.x] = 0u;
    __syncthreads();
    int e = blockIdx.x * blockDim.x + threadIdx.x;
    if (e < E) {
        int row = ei[e], col = ei[E + e];
        const float* sr = ssrc + (size_t)row * HEADS;
        const float* sd = sdst + (size_t)col * HEADS;
        #pragma unroll
        for (int h = 0; h < HEADS; ++h)
            atomicMax(&sm[h], fmap(lrelu(sr[h] + sd[h])));
    }
    __syncthreads();
    if (threadIdx.x < HEADS) atomicMax(&maxKeys[threadIdx.x], sm[threadIdx.x]);
}

// ---- pass 2: per-head exp-sum ----
__global__ void k_edge_sum(const int* __restrict__ ei, const float* __restrict__ ssrc,
                           const float* __restrict__ sdst, const unsigned* __restrict__ maxKeys,
                           float* __restrict__ sums, int E) {
    __shared__ float sacc[HEADS];
    __shared__ float smax[HEADS];
    if (threadIdx.x < HEADS) { sacc[threadIdx.x] = 0.f; smax[threadIdx.x] = funmap(maxKeys[threadIdx.x]); }
    __syncthreads();
    int e = blockIdx.x * blockDim.x + threadIdx.x;
    if (e < E) {
        int row = ei[e], col = ei[E + e];
        const float* sr = ssrc + (size_t)row * HEADS;
        const float* sd = sdst + (size_t)col * HEADS;
        #pragma unroll
        for (int h = 0; h < HEADS; ++h)
            atomicAdd(&sacc[h], __expf(lrelu(sr[h] + sd[h]) - smax[h]));
    }
    __syncthreads();
    if (threadIdx.x < HEADS) atomicAdd(&sums[threadIdx.x], sacc[threadIdx.x]);
}

// ---- pass 3: gather + head-weighted combine + scatter-add (1 wave / edge) ----
__global__ void __launch_bounds__(256)
k_scatter(const int* __restrict__ ei, const float* __restrict__ xt,
          const float* __restrict__ ssrc, const float* __restrict__ sdst,
          const unsigned* __restrict__ maxKeys, const float* __restrict__ sums,
          const float* __restrict__ dwH, float* __restrict__ out, int E) {
    const int wave = threadIdx.x >> 5;
    const int lane = threadIdx.x & 31;
    const int e = blockIdx.x * 8 + wave;
    if (e >= E) return;
    const int row = ei[e];
    const int col = ei[E + e];

    // prefetch next edge's 2KB gather block (32 lanes x 64B stride)
    const int en = e + 8;
    if (en < E) {
        int cn = ei[E + en];
        __builtin_prefetch(xt + (size_t)cn * HO + lane * 16, 0, 0);
    }

    float w = 0.f;
    if (lane < HEADS) {
        float v = lrelu(ssrc[(size_t)row * HEADS + lane] + sdst[(size_t)col * HEADS + lane]);
        w = __expf(v - funmap(maxKeys[lane])) / sums[lane] * dwH[lane];
    }
    const float* base = xt + (size_t)col * HO;
    float a0 = 0.f, a1 = 0.f;
    #pragma unroll
    for (int h = 0; h < HEADS; ++h) {
        float wh = __shfl(w, h, 32);
        a0 += wh * base[h * OUT_C + lane];
        a1 += wh * base[h * OUT_C + lane + 32];
    }
    atomicAdd(&out[(size_t)row * OUT_C + lane],      a0);
    atomicAdd(&out[(size_t)row * OUT_C + lane + 32], a1);
}

extern "C" void kernel_launch(void* const* d_in, const int* in_sizes, int n_in,
                              void* d_out, int out_size, void* d_ws, size_t ws_size,
                              hipStream_t stream) {
    const float* x    = (const float*)d_in[0];
    const int*   ei   = (const int*)  d_in[1];
    const float* dc   = (const float*)d_in[2];
    const float* mw   = (const float*)d_in[3];
    const float* matt = (const float*)d_in[4];
    const float* dW   = (const float*)d_in[5];
    const float* db   = (const float*)d_in[6];
    const float* dA   = (const float*)d_in[7];
    float* out = (float*)d_out;

    const int n = in_sizes[0] / IN_C;
    const int E = in_sizes[1] / 2;

    // workspace layout (floats): dwH[8] | maxKeys[8] | sums[8] | pad | ssrc[n*8] | sdst[n*8] | xt[n*512]
    float*    ws      = (float*)d_ws;
    float*    dwH     = ws;
    unsigned* maxKeys = (unsigned*)(ws + 8);
    float*    sums    = ws + 16;
    float*    ssrc    = ws + 32;
    float*    sdst    = ssrc + (size_t)n * HEADS;
    float*    xt      = sdst + (size_t)n * HEADS;

    k_init   <<<2048, 256, 0, stream>>>(out, (long long)n * OUT_C, maxKeys, sums);
    k_domain <<<1, OUT_C, 0, stream>>>(dc, dW, db, dA, dwH);
    k_gemm   <<<(n + 15) / 16, 256, 0, stream>>>(x, mw, matt, xt, ssrc, sdst, n);
    k_edge_max<<<(E + 255) / 256, 256, 0, stream>>>(ei, ssrc, sdst, maxKeys, E);
    k_edge_sum<<<(E + 255) / 256, 256, 0, stream>>>(ei, ssrc, sdst, maxKeys, sums, E);
    k_scatter<<<(E + 7) / 8, 256, 0, stream>>>(ei, xt, ssrc, sdst, maxKeys, sums, dwH, out, E);
}